// PiecewiseLinearShapeNN2D_24816321036907
// MI455X (gfx1250) — compile-verified
//
#include <hip/hip_runtime.h>
#include <math.h>

#define ALPHA_RBF 0.2f
#define M_EVAL 16384
#define NPATCH 24
#define NSYS 32
#define NAUG 35   /* 32 G columns + 3 RHS columns */

typedef float v2f __attribute__((ext_vector_type(2)));
typedef float v8f __attribute__((ext_vector_type(8)));

__global__ __launch_bounds__(96) void mls_shape_kernel(
    const float* __restrict__ x_eval,       // (M,2)
    const int*   __restrict__ elem_id,      // (M,)
    const float* __restrict__ node_coords,  // (NN,2)
    const float* __restrict__ u_full,       // (NN,2)
    const int*   __restrict__ connectivity, // (NE,3)
    const int*   __restrict__ patch,        // (NN,24)
    float*       __restrict__ out)          // u_h(M,2) | detJ(M) | grad_u(M,2,2)
{
  __shared__ float sCx[3][32], sCy[3][32], sPm[3][32];
  __shared__ float sUx[3][32], sUy[3][32], sNm[3][32];
  __shared__ float sG[3][32*33];
  __shared__ float sRes[3][8];

  const int m    = blockIdx.x;
  const int wave = threadIdx.x >> 5;   // element-node index i = 0..2
  const int lane = threadIdx.x & 31;

  // ---- element, shape functions, physical point (uniform per block) ----
  const int e  = elem_id[m];
  const int n0 = connectivity[3*e+0];
  const int n1 = connectivity[3*e+1];
  const int n2 = connectivity[3*e+2];
  const float xi  = x_eval[2*m+0];
  const float eta = x_eval[2*m+1];
  const float Ns0 = xi, Ns1 = eta, Ns2 = 1.0f - xi - eta;
  const float ex0 = node_coords[2*n0+0], ey0 = node_coords[2*n0+1];
  const float ex1 = node_coords[2*n1+0], ey1 = node_coords[2*n1+1];
  const float ex2 = node_coords[2*n2+0], ey2 = node_coords[2*n2+1];
  const float xp = Ns0*ex0 + Ns1*ex1 + Ns2*ex2;
  const float yp = Ns0*ey0 + Ns1*ey1 + Ns2*ey2;

  const int ni = (wave==0) ? n0 : (wave==1 ? n1 : n2);

  // ---- per-lane patch entry (lanes 0..23 real, 24..31 padding) ----
  float cx=0.f, cy=0.f, pm=0.f, ux=0.f, uy=0.f;
  if (lane < NPATCH) {
    int pj    = patch[ni*NPATCH + lane];
    int valid = (pj >= 0);
    int ps    = valid ? pj : 0;
    pm = valid ? 1.0f : 0.0f;
    cx = node_coords[2*ps+0];
    cy = node_coords[2*ps+1];
    ux = pm * u_full[2*ps+0];
    uy = pm * u_full[2*ps+1];
  }
  sCx[wave][lane] = cx;  sCy[wave][lane] = cy;  sPm[wave][lane] = pm;
  sUx[wave][lane] = ux;  sUy[wave][lane] = uy;
  sNm[wave][lane] = cx*cx + cy*cy;
  __syncthreads();

  // ---- RBF 24x24 (padded 32x32) Gram block via V_WMMA_F32_16X16X4_F32 ----
  float* G = sG[wave];
  const int half = lane >> 4;   // 0: K=0,1  1: K=2,3 (zero pad)
  const int l15  = lane & 15;
  #pragma unroll
  for (int ti = 0; ti < 2; ++ti) {
    v2f A;
    A.x = (half==0) ? sCx[wave][16*ti + l15] : 0.0f;  // K=0 : x
    A.y = (half==0) ? sCy[wave][16*ti + l15] : 0.0f;  // K=1 : y
    #pragma unroll
    for (int tj = 0; tj < 2; ++tj) {
      v2f B;
      B.x = (half==0) ? sCx[wave][16*tj + l15] : 0.0f;
      B.y = (half==0) ? sCy[wave][16*tj + l15] : 0.0f;
      v8f d = {0.f,0.f,0.f,0.f,0.f,0.f,0.f,0.f};
      d = __builtin_amdgcn_wmma_f32_16x16x4_f32(false, A, false, B,
                                                (short)0, d, false, false);
      #pragma unroll
      for (int v = 0; v < 8; ++v) {
        int row = 16*ti + v + 8*half;
        int col = 16*tj + l15;
        float r2 = fmaxf(sNm[wave][row] + sNm[wave][col] - 2.0f*d[v], 0.0f);
        float r  = sqrtf(r2);
        float s  = r * (1.0f/ALPHA_RBF);
        float om = 1.0f - s;
        float val = (s <= 1.0f)
                  ? (om*om*(1.0f + 2.0f*s) * sPm[wave][row] * sPm[wave][col])
                  : 0.0f;
        G[row*33 + col] = val;
      }
    }
  }

  // ---- polynomial border, zero block, identity padding, diag regularizer ----
  if (lane < NPATCH) {
    float pv[6] = {1.0f, cx, cy, cx*cx, cx*cy, cy*cy};
    #pragma unroll
    for (int q = 0; q < 6; ++q) {
      G[lane*33 + 24+q]   = pm*pv[q];
      G[(24+q)*33 + lane] = pm*pv[q];
    }
    G[lane*33 + 30] = 0.0f;
    G[lane*33 + 31] = 0.0f;
    G[lane*33 + lane] += 1e-12f*(1.0f - pm);
  } else {
    #pragma unroll
    for (int q = 24; q < 32; ++q) G[lane*33 + q] = 0.0f;
    if (lane == 30) G[30*33+30] = 1.0f;
    if (lane == 31) G[31*33+31] = 1.0f;
  }
  __syncthreads();

  // ---- load my row + build RHS (value, d/dx, d/dy) ----
  float a[NAUG];
  #pragma unroll
  for (int c = 0; c < 32; ++c) a[c] = G[lane*33 + c];
  {
    float b0=0.f, b1=0.f, b2=0.f;
    if (lane < NPATCH) {
      float dx = xp - cx, dy = yp - cy;
      float rr = sqrtf(dx*dx + dy*dy);
      float s  = rr * (1.0f/ALPHA_RBF);
      float mk = ((s <= 1.0f) ? 1.0f : 0.0f) * pm;
      float om = 1.0f - s;
      b0 = mk * om*om*(1.0f + 2.0f*s);
      float dRdr = mk * (-6.0f) * om * (1.0f/ALPHA_RBF);
      float c0 = dRdr / (rr + 1e-12f);
      b1 = c0*dx;  b2 = c0*dy;
    } else if (lane < 30) {
      int k = lane - 24;
      if      (k==0) { b0=1.0f;  b1=0.0f;     b2=0.0f;     }
      else if (k==1) { b0=xp;    b1=1.0f;     b2=0.0f;     }
      else if (k==2) { b0=yp;    b1=0.0f;     b2=1.0f;     }
      else if (k==3) { b0=xp*xp; b1=2.0f*xp;  b2=0.0f;     }
      else if (k==4) { b0=xp*yp; b1=yp;       b2=xp;       }
      else           { b0=yp*yp; b1=0.0f;     b2=2.0f*yp;  }
    }
    a[32]=b0; a[33]=b1; a[34]=b2;
  }

  // ---- Gauss-Jordan with partial pivoting, row-per-lane ----
  bool pivoted = false;
  int  colOf   = 0;
  float diag   = 1.0f;
  #pragma unroll
  for (int k = 0; k < NSYS; ++k) {
    float cur = a[k];
    float mag = pivoted ? -1.0f : fabsf(cur);
    int   bl  = lane;
    #pragma unroll
    for (int off = 16; off > 0; off >>= 1) {
      float omv = __shfl_xor(mag, off, 32);
      int   ol  = __shfl_xor(bl,  off, 32);
      if (omv > mag || (omv == mag && ol < bl)) { mag = omv; bl = ol; }
    }
    const int p   = bl;                    // uniform pivot lane
    const float pv = __shfl(cur, p, 32);
    const float f  = (lane == p) ? 0.0f : (cur / pv);
    #pragma unroll
    for (int c = k; c < NAUG; ++c) {       // cols < k already zero in pivot row
      float bc = __shfl(a[c], p, 32);
      a[c] -= f * bc;
    }
    if (lane == p) { pivoted = true; colOf = k; diag = pv; }
  }

  // ---- extract W / dW for my unknown, contract with u_patch ----
  const float invd = 1.0f / diag;
  const float W   = a[32]*invd;
  const float dwx = a[33]*invd;
  const float dwy = a[34]*invd;
  float uxj = 0.f, uyj = 0.f;
  if (colOf < NPATCH) { uxj = sUx[wave][colOf]; uyj = sUy[wave][colOf]; }
  float wux = W*uxj, wuy = W*uyj;
  float g00 = uxj*dwx, g01 = uxj*dwy, g10 = uyj*dwx, g11 = uyj*dwy;
  #pragma unroll
  for (int off = 16; off > 0; off >>= 1) {
    wux += __shfl_xor(wux, off, 32);
    wuy += __shfl_xor(wuy, off, 32);
    g00 += __shfl_xor(g00, off, 32);
    g01 += __shfl_xor(g01, off, 32);
    g10 += __shfl_xor(g10, off, 32);
    g11 += __shfl_xor(g11, off, 32);
  }
  if (lane == 0) {
    sRes[wave][0]=wux; sRes[wave][1]=wuy;
    sRes[wave][2]=g00; sRes[wave][3]=g01;
    sRes[wave][4]=g10; sRes[wave][5]=g11;
  }
  __syncthreads();

  // ---- combine per eval point ----
  if (threadIdx.x == 0) {
    float uhx = Ns0*sRes[0][0] + Ns1*sRes[1][0] + Ns2*sRes[2][0];
    float uhy = Ns0*sRes[0][1] + Ns1*sRes[1][1] + Ns2*sRes[2][1];
    float j00 = ex0-ex2, j01 = ex1-ex2, j10 = ey0-ey2, j11 = ey1-ey2;
    float det  = j00*j11 - j01*j10;
    float idet = 1.0f/det;
    float i00 =  j11*idet, i01 = -j01*idet, i10 = -j10*idet, i11 = j00*idet;
    // dN_dx[a][j]: col0=Jinv[:,0], col1=Jinv[:,1], col2=-(col0+col1)
    float dN[2][3] = { { i00, i01, -(i00+i01) },
                       { i10, i11, -(i10+i11) } };
    float gout[2][2];
    #pragma unroll
    for (int c2 = 0; c2 < 2; ++c2)
      #pragma unroll
      for (int k2 = 0; k2 < 2; ++k2) {
        float s1 = 0.0f;                       // grad1[c][k] = sum_j dN[k][j]*Wu[j][c]
        #pragma unroll
        for (int j = 0; j < 3; ++j) s1 += dN[k2][j]*sRes[j][c2];
        float s2 = sRes[0][2+c2*2+k2] + sRes[1][2+c2*2+k2] + sRes[2][2+c2*2+k2];
        gout[c2][k2] = s1 + s2;
      }
    out[2*m+0] = uhx;
    out[2*m+1] = uhy;
    out[2*M_EVAL + m] = det;
    out[3*M_EVAL + 4*m + 0] = gout[0][0];
    out[3*M_EVAL + 4*m + 1] = gout[0][1];
    out[3*M_EVAL + 4*m + 2] = gout[1][0];
    out[3*M_EVAL + 4*m + 3] = gout[1][1];
  }
}

extern "C" void kernel_launch(void* const* d_in, const int* in_sizes, int n_in,
                              void* d_out, int out_size, void* d_ws, size_t ws_size,
                              hipStream_t stream) {
  (void)in_sizes; (void)n_in; (void)out_size; (void)d_ws; (void)ws_size;
  const float* x_eval       = (const float*)d_in[0];
  const int*   elem_id      = (const int*)  d_in[1];
  const float* node_coords  = (const float*)d_in[2];
  const float* u_full       = (const float*)d_in[3];
  const int*   connectivity = (const int*)  d_in[4];
  const int*   patch        = (const int*)  d_in[5];
  float* out = (float*)d_out;
  hipLaunchKernelGGL(mls_shape_kernel, dim3(M_EVAL), dim3(96), 0, stream,
                     x_eval, elem_id, node_coords, u_full, connectivity, patch, out);
}